// ConvM2Cartesian_18880676233724
// MI455X (gfx1250) — compile-verified
//
#include <hip/hip_runtime.h>
#include <cstddef>
#include <cstdint>

#define IN_CH  32
#define OUT_CH 32
#define ORIENT 8
#define HH 128
#define WW 128
#define HO 124
#define WO 124

// LDS input patch: 12 rows x 36 cols x 32 cin (cin padded to 34, 8B-aligned rows)
#define LXR 12
#define LXC 36
#define XCP 34
// LDS weight slice: 5 dx x 32 cout x 32 cin (cin padded to 36 -> 16B-aligned rows)
#define WCP 36
#define XBUF (LXR * LXC * XCP)   // 14688 floats per buffer
#define WBUF (5 * OUT_CH * WCP)  //  5760 floats per buffer

typedef __attribute__((ext_vector_type(2))) float v2f;
typedef __attribute__((ext_vector_type(8))) float v8f;
typedef __attribute__((ext_vector_type(4))) int   v4i;

#define AS1 __attribute__((address_space(1)))
#define AS3 __attribute__((address_space(3)))

// ---- CDNA5 async global<->LDS helpers (builtin if present, else inline asm) ----
__device__ __forceinline__ void async_g2l_b32(const float* g, float* l) {
#if __has_builtin(__builtin_amdgcn_global_load_async_to_lds_b32)
  __builtin_amdgcn_global_load_async_to_lds_b32(
      (AS1 int*)g, (AS3 int*)l, 0, 0);
#else
  asm volatile("global_load_async_to_lds_b32 %0, %1, off"
               :: "v"((uint32_t)(uintptr_t)l), "v"(g) : "memory");
#endif
}

__device__ __forceinline__ void async_g2l_b128(const float* g, float* l) {
#if __has_builtin(__builtin_amdgcn_global_load_async_to_lds_b128)
  __builtin_amdgcn_global_load_async_to_lds_b128(
      (AS1 v4i*)g, (AS3 v4i*)l, 0, 0);
#else
  asm volatile("global_load_async_to_lds_b128 %0, %1, off"
               :: "v"((uint32_t)(uintptr_t)l), "v"(g) : "memory");
#endif
}

__device__ __forceinline__ void async_l2g_b128(float* g, const float* l) {
#if __has_builtin(__builtin_amdgcn_global_store_async_from_lds_b128)
  __builtin_amdgcn_global_store_async_from_lds_b128(
      (AS1 v4i*)g, (AS3 v4i*)l, 0, 0);
#else
  asm volatile("global_store_async_from_lds_b128 %0, %1, off"
               :: "v"(g), "v"((uint32_t)(uintptr_t)l) : "memory");
#endif
}

#if __has_builtin(__builtin_amdgcn_s_wait_asynccnt)
#define ASYNC_WAIT(n) __builtin_amdgcn_s_wait_asynccnt(n)
#else
#define ASYNC_WAIT(n) asm volatile("s_wait_asynccnt %0" :: "i"(n) : "memory")
#endif

// ---------------------------------------------------------------------------
// Stage 1: build kernel stack ks[i][a][dy][dx][cout][cin] (8*125*32*32 floats)
// ---------------------------------------------------------------------------
__device__ __forceinline__ float bspline2(float t) {
  float a = fabsf(t);
  float u = 1.5f - a;
  float r = (a < 1.5f) ? 0.5f * u * u : 0.0f;
  if (a <= 0.5f) r = 0.75f - a * a;
  return r;
}

__global__ void build_ks(const float* __restrict__ wgt, float* __restrict__ ks) {
  __shared__ float bas[128];
  const int bid = blockIdx.x;          // 0..999 == io*125 + (a*25 + dy*5 + dx)
  const int io  = bid / 125;
  const int pos = bid % 125;
  const int a   = pos / 25;
  const int dy  = (pos / 5) % 5;
  const int dx  = pos % 5;
  const int t   = threadIdx.x;

  if (t < 125) {
    const int n0 = t / 25, n1 = (t / 5) % 5, n2 = t % 5;
    const float ang = -(float)io * 0.78539816339744830962f;  // -io * 2*pi/8
    const float cs = cosf(ang), sn = sinf(ang);
    const float xg = (float)(dx - 2), yg = (float)(dy - 2);
    const float xr = xg * cs + yg * sn;
    const float yr = -xg * sn + yg * cs;
    bas[t] = bspline2((float)(a - n0)) *
             bspline2(yr - (float)(n1 - 2)) *
             bspline2(xr - (float)(n2 - 2));
  }
  __syncthreads();

  for (int p = t; p < OUT_CH * IN_CH; p += 256) {
    const float* wp = wgt + (size_t)p * 125;   // weights[cout][cin][125]
    float acc = 0.f;
#pragma unroll 5
    for (int n = 0; n < 125; ++n) acc = fmaf(wp[n], bas[n], acc);
    const int cout = p >> 5, cin = p & 31;
    ks[((size_t)bid * OUT_CH + cout) * IN_CH + cin] = acc;
  }
}

// ---------------------------------------------------------------------------
// Stage 2: implicit-GEMM conv on V_WMMA_F32_16X16X4_F32 with double-buffered
// GLOBAL_LOAD_ASYNC_TO_LDS staging.
// Grid: (4 x-tiles, 16 y-tiles, 64 = b*8+io). Block: 256 threads = 8 waves.
// Tile: 8 rows x 32 cols x 32 couts; wave w owns row w = 2x2 WMMA subtiles.
// ---------------------------------------------------------------------------
__global__ __launch_bounds__(256)
void conv_m2(const float* __restrict__ x, const float* __restrict__ ks,
             float* __restrict__ out) {
  __shared__ __align__(16) float lds_x[2 * XBUF];  // 117504 B
  __shared__ __align__(16) float lds_w[2 * WBUF];  //  46080 B

  const int tid   = threadIdx.x;
  const int lane  = tid & 31;
  const int wv    = tid >> 5;      // wave id = output row within tile
  const int mlane = lane & 15;
  const int hi    = lane >> 4;
  const int kk    = hi * 2;        // k pair base within a K=4 chunk

  const int x0 = blockIdx.x * 32;
  const int y0 = blockIdx.y * 8;
  const int b  = blockIdx.z >> 3;
  const int io = blockIdx.z & 7;

  // ---- async staging issue helpers (uniform op count per thread!) ----
  // x stage: 13824 = 54*256 b32 ops; w stage: 1280 = 5*256 b128 ops.
  auto issue_x = [&](int a_, float* buf) {
    const int ori = (io + a_ + 6) & 7;  // (io + a - 2) mod 8, wrap padding
    const float* xsrc =
        x + ((size_t)b * IN_CH * ORIENT + ori) * (size_t)(HH * WW);
    for (int e = tid; e < LXR * LXC * IN_CH; e += 256) {
      const int c   = e % LXC;
      const int r   = (e / LXC) % LXR;
      const int cin = e / (LXC * LXR);
      int yr = y0 + r; if (yr > HH - 1) yr = HH - 1;  // clamp: keep EXEC full,
      int xc = x0 + c; if (xc > WW - 1) xc = WW - 1;  // OOB feeds dropped outputs
      async_g2l_b32(xsrc + (size_t)cin * (ORIENT * HH * WW) + yr * WW + xc,
                    buf + (r * LXC + c) * XCP + cin);
    }
  };
  auto issue_w = [&](int s_, float* buf) {  // s_ = a*5+dy
    const float* wsrc = ks + (size_t)(io * 125 + s_ * 5) * (OUT_CH * IN_CH);
    for (int u = tid; u < (5 * OUT_CH * IN_CH) / 4; u += 256) {
      const int cin4 = u & 7, cout = (u >> 3) & 31, dxp = u >> 8;
      async_g2l_b128(wsrc + (size_t)u * 4,
                     buf + (dxp * OUT_CH + cout) * WCP + cin4 * 4);
    }
  };

  v8f acc00 = {0.f, 0.f, 0.f, 0.f, 0.f, 0.f, 0.f, 0.f};
  v8f acc01 = acc00, acc10 = acc00, acc11 = acc00;

  // Per-lane fragment bases (floats). A 16x4 f32 ISA layout:
  // lanes 0-15 hold k={0,1}, lanes 16-31 k={2,3}, M = lane%16. B analogous.
  const int aB0 = (wv * LXC + mlane) * XCP + kk;  // cols 0..15
  const int aB1 = aB0 + 16 * XCP;                 // cols 16..31
  const int bB0 = mlane * WCP + kk;               // couts 0..15
  const int bB1 = bB0 + 16 * WCP;                 // couts 16..31

  // ---- prologue: stage 0 ----
  issue_x(0, lds_x);
  issue_w(0, lds_w);

  for (int a = 0; a < 5; ++a) {
    const float* xbuf = lds_x + (a & 1) * XBUF;
    for (int dy = 0; dy < 5; ++dy) {
      const int s = a * 5 + dy;
      // issue next stage into the idle buffers (order matters for ASYNCcnt):
      if (s + 1 < 25) issue_w(s + 1, lds_w + ((s + 1) & 1) * WBUF);
      if (dy == 0 && a < 4) issue_x(a + 1, lds_x + ((a + 1) & 1) * XBUF);
      // retire this stage's copies; leave the younger prefetches in flight.
      if (a < 4 && dy < 2) ASYNC_WAIT(59);   // 5 (w) + 54 (x) younger ops
      else if (s < 24)     ASYNC_WAIT(5);    // 5 (w) younger ops
      else                 ASYNC_WAIT(0);
      __syncthreads();

      const float* wbuf = lds_w + (s & 1) * WBUF;
      for (int dx = 0; dx < 5; ++dx) {
        const int aoff = (dy * LXC + dx) * XCP;
        const int boff = dx * (OUT_CH * WCP);
#pragma unroll
        for (int c4 = 0; c4 < 8; ++c4) {
          const int ka = c4 * 4;
          const v2f a0 = *(const v2f*)(xbuf + aB0 + aoff + ka);
          const v2f a1 = *(const v2f*)(xbuf + aB1 + aoff + ka);
          const v2f b0 = *(const v2f*)(wbuf + bB0 + boff + ka);
          const v2f b1 = *(const v2f*)(wbuf + bB1 + boff + ka);
          acc00 = __builtin_amdgcn_wmma_f32_16x16x4_f32(
              false, a0, false, b0, (short)0, acc00, false, false);
          acc01 = __builtin_amdgcn_wmma_f32_16x16x4_f32(
              false, a0, false, b1, (short)0, acc01, false, false);
          acc10 = __builtin_amdgcn_wmma_f32_16x16x4_f32(
              false, a1, false, b0, (short)0, acc10, false, false);
          acc11 = __builtin_amdgcn_wmma_f32_16x16x4_f32(
              false, a1, false, b1, (short)0, acc11, false, false);
        }
      }
      __syncthreads();
    }
  }

  // ---- transpose accumulators through LDS, then async store b128 ----
  float* lout = lds_x;  // reuse: [cout][row 8][col pad 36] = 9216 floats
#pragma unroll
  for (int vi = 0; vi < 8; ++vi) {
    const int mc = vi + 8 * hi;  // M within 16x16 subtile (C/D VGPR layout)
    lout[((mlane)      * 8 + wv) * 36 + mc]      = acc00[vi];
    lout[((mlane + 16) * 8 + wv) * 36 + mc]      = acc01[vi];
    lout[((mlane)      * 8 + wv) * 36 + 16 + mc] = acc10[vi];
    lout[((mlane + 16) * 8 + wv) * 36 + 16 + mc] = acc11[vi];
  }
  __syncthreads();

  float* outbase =
      out + (((size_t)b * OUT_CH) * ORIENT + io) * (size_t)(HO * WO);
  for (int u = tid; u < OUT_CH * 8 * 8; u += 256) {  // 2048 x 4-float chunks
    const int col = (u & 7) * 4, row = (u >> 3) & 7, cout = u >> 6;
    const int yy = y0 + row, xx = x0 + col;
    if (yy < HO && xx < WO)  // chunks are whole-valid or whole-invalid in x
      async_l2g_b128(outbase + (size_t)cout * (ORIENT * HO * WO) +
                         (size_t)yy * WO + xx,
                     lout + (cout * 8 + row) * 36 + col);
  }
  // S_ENDPGM performs an implicit wait-idle, retiring the async stores.
}

// ---------------------------------------------------------------------------
extern "C" void kernel_launch(void* const* d_in, const int* in_sizes, int n_in,
                              void* d_out, int out_size, void* d_ws,
                              size_t ws_size, hipStream_t stream) {
  (void)in_sizes; (void)n_in; (void)out_size; (void)ws_size;
  const float* x   = (const float*)d_in[0];   // [8,32,8,128,128] f32
  const float* wgt = (const float*)d_in[1];   // [32,32,125] f32
  float* out = (float*)d_out;                 // [8,32,8,124,124] f32
  float* ks  = (float*)d_ws;                  // 8*125*32*32 floats = 4 MB

  build_ks<<<dim3(1000), dim3(256), 0, stream>>>(wgt, ks);
  conv_m2<<<dim3(4, 16, 64), dim3(256), 0, stream>>>(x, ks, out);
}